// MixHop_7473243095280
// MI455X (gfx1250) — compile-verified
//
#include <hip/hip_runtime.h>

// MixHop GCN for MI455X (gfx1250).
// Propagation (scatter-add over 850K edges, fp32 atomics resolving in 192MB L2)
// is the bandwidth-bound bulk; GEMMs (~9 GFLOP total) use fp32 WMMA
// (v_wmma_f32_16x16x4_f32) to keep full precision on the matrix pipe.

#define F_IN  128
#define HID   60
#define CATF  180
#define NCLS  7
#define EPSB  1e-5f
#define KMAX  192   // LDS W-tile K capacity (>= 180)

typedef float v2f __attribute__((ext_vector_type(2)));
typedef float v8f __attribute__((ext_vector_type(8)));

// ---------------- utility kernels ----------------

__global__ void fill_kernel(float* __restrict__ p, float v, long n) {
    long i = (long)blockIdx.x * blockDim.x + threadIdx.x;
    long st = (long)gridDim.x * blockDim.x;
    for (; i < n; i += st) p[i] = v;
}

__global__ void deg_kernel(const int* __restrict__ dst, float* __restrict__ deg, int E) {
    int i = blockIdx.x * blockDim.x + threadIdx.x;
    int st = gridDim.x * blockDim.x;
    for (; i < E; i += st) atomicAdd(&deg[dst[i]], 1.0f);
}

__global__ void rsqrt_kernel(float* __restrict__ d, int n) {
    int i = blockIdx.x * blockDim.x + threadIdx.x;
    int st = gridDim.x * blockDim.x;
    for (; i < n; i += st) d[i] = rsqrtf(d[i]);   // deg >= 1 (self loop) always
}

// ---------------- graph propagation: out[dst] += dinv[src]*dinv[dst]*x[src] ----------------
// One wave per edge (or self-loop for e >= E); 32 lanes stripe the feature dim.

__global__ void prop_kernel(const float* __restrict__ x,
                            const int* __restrict__ esrc,
                            const int* __restrict__ edst,
                            const float* __restrict__ dinv,
                            float* __restrict__ out,
                            int F, int E, int N) {
    int gtid = blockIdx.x * blockDim.x + threadIdx.x;
    int wid  = gtid >> 5;
    int lane = threadIdx.x & 31;
    int nw   = (gridDim.x * blockDim.x) >> 5;
    int total = E + N;
    for (int e = wid; e < total; e += nw) {
        int s, d;
        if (e < E) { s = esrc[e]; d = edst[e]; }
        else       { s = d = e - E; }
        float nrm = dinv[s] * dinv[d];
        const float* xs = x + (size_t)s * F;
        float* od = out + (size_t)d * F;
        for (int f = lane; f < F; f += 32)
            atomicAdd(&od[f], nrm * xs[f]);
    }
}

// ---------------- fp32 WMMA GEMM: OUT[:, coff:coff+ncols] = X[N,K] @ W[K,ncols] + bias ----------------
// 4 waves / block; each wave computes one 16x16 tile via v_wmma_f32_16x16x4_f32.

__global__ void gemm_wmma_kernel(const float* __restrict__ X, int ldx, int K,
                                 const float* __restrict__ W,
                                 const float* __restrict__ bias, int ncols,
                                 float* __restrict__ OUT, int ldout, int coff,
                                 int nrows) {
    __shared__ float lw[KMAX * 16];
    int n0 = blockIdx.y * 16;

    // Stage W column tile (zero-padded past ncols) in LDS.
    for (int idx = threadIdx.x; idx < K * 16; idx += blockDim.x) {
        int k = idx >> 4, n = idx & 15, c = n0 + n;
        lw[idx] = (c < ncols) ? W[(size_t)k * ncols + c] : 0.0f;
    }
    __syncthreads();

    int lane = threadIdx.x & 31;
    int wave = threadIdx.x >> 5;
    int row0 = (blockIdx.x * 4 + wave) * 16;
    if (row0 >= nrows) return;               // wave-uniform: EXEC stays all-ones

    int half = lane >> 4;                    // 0: K=k,k+1 ; 1: K=k+2,k+3
    int nn   = lane & 15;

    v8f acc = {0.f, 0.f, 0.f, 0.f, 0.f, 0.f, 0.f, 0.f};
    const float* xrow = X + (size_t)(row0 + nn) * ldx + 2 * half;

    for (int k = 0; k < K; k += 4) {
        v2f a, b;
        // A 16x4 layout: lane(0-15)=M, VGPR0/1 = K=k+2*half, k+2*half+1
        a.x = xrow[k];
        a.y = xrow[k + 1];
        int kk = k + 2 * half;
        // B 4x16 layout: lane(0-15)=N, VGPR0/1 = K rows kk, kk+1
        b.x = lw[kk * 16 + nn];
        b.y = lw[kk * 16 + 16 + nn];
        acc = __builtin_amdgcn_wmma_f32_16x16x4_f32(
            false, a, false, b, (short)0, acc, false, false);
    }

    int col = n0 + nn;
    if (col < ncols) {
        float bv = bias[col];
#pragma unroll
        for (int v = 0; v < 8; ++v) {
            int m = row0 + v + 8 * half;     // D: lanes 0-15 -> M=v, lanes 16-31 -> M=v+8
            OUT[(size_t)m * ldout + coff + col] = acc[v] + bv;
        }
    }
}

// ---------------- batchnorm (training-mode stats over the node dim) ----------------

__global__ void bn_stats_kernel(const float* __restrict__ h,
                                float* __restrict__ stats,  // [2*CATF]: sum, sumsq
                                int n, int cat) {
    __shared__ float ssum[CATF];
    __shared__ float ssq[CATF];
    for (int i = threadIdx.x; i < cat; i += blockDim.x) { ssum[i] = 0.f; ssq[i] = 0.f; }
    __syncthreads();
    size_t total = (size_t)n * cat;
    size_t st = (size_t)gridDim.x * blockDim.x;
    for (size_t i = (size_t)blockIdx.x * blockDim.x + threadIdx.x; i < total; i += st) {
        float v = h[i];
        int c = (int)(i % cat);
        atomicAdd(&ssum[c], v);
        atomicAdd(&ssq[c], v * v);
    }
    __syncthreads();
    for (int i = threadIdx.x; i < cat; i += blockDim.x) {
        atomicAdd(&stats[i], ssum[i]);
        atomicAdd(&stats[cat + i], ssq[i]);
    }
}

__global__ void bn_apply_kernel(float* __restrict__ h,
                                const float* __restrict__ stats,
                                const float* __restrict__ g,
                                const float* __restrict__ b,
                                int n, int cat) {
    size_t total = (size_t)n * cat;
    size_t st = (size_t)gridDim.x * blockDim.x;
    float inv_n = 1.0f / (float)n;
    for (size_t i = (size_t)blockIdx.x * blockDim.x + threadIdx.x; i < total; i += st) {
        int c = (int)(i % cat);
        float mean = stats[c] * inv_n;
        float var  = stats[cat + c] * inv_n - mean * mean;   // biased variance
        float scl  = g[c] * rsqrtf(var + EPSB);
        h[i] = (h[i] - mean) * scl + b[c];
    }
}

// ---------------- host orchestration ----------------

extern "C" void kernel_launch(void* const* d_in, const int* in_sizes, int n_in,
                              void* d_out, int out_size, void* d_ws, size_t ws_size,
                              hipStream_t stream) {
    (void)n_in; (void)out_size; (void)ws_size;

    const float* x  = (const float*)d_in[0];
    const int*   ei = (const int*)d_in[1];   // [2,E] flat: row0 = src, row1 = dst
    // setup_inputs() dict order: per p in 0..2: c1_w,c1_b,c2_w,c2_b,c3_w,c3_b
    const float* c1_w[3] = {(const float*)d_in[2],  (const float*)d_in[8],  (const float*)d_in[14]};
    const float* c1_b[3] = {(const float*)d_in[3],  (const float*)d_in[9],  (const float*)d_in[15]};
    const float* c2_w[3] = {(const float*)d_in[4],  (const float*)d_in[10], (const float*)d_in[16]};
    const float* c2_b[3] = {(const float*)d_in[5],  (const float*)d_in[11], (const float*)d_in[17]};
    const float* c3_w[3] = {(const float*)d_in[6],  (const float*)d_in[12], (const float*)d_in[18]};
    const float* c3_b[3] = {(const float*)d_in[7],  (const float*)d_in[13], (const float*)d_in[19]};
    const float* bn_g[3] = {(const float*)d_in[20], (const float*)d_in[22], (const float*)d_in[24]};
    const float* bn_b[3] = {(const float*)d_in[21], (const float*)d_in[23], (const float*)d_in[25]};
    const float* lin_w = (const float*)d_in[26];
    const float* lin_b = (const float*)d_in[27];
    float* out = (float*)d_out;

    const int N = in_sizes[0] / F_IN;
    const int E = in_sizes[1] / 2;
    const int* esrc = ei;
    const int* edst = ei + E;

    // Workspace carve-out (~145 MB): 4 x [N,CAT] ping-pong buffers + dinv + stats.
    float* ws    = (float*)d_ws;
    size_t NC    = (size_t)N * CATF;
    float* buf0  = ws;
    float* buf1  = buf0 + NC;
    float* buf2  = buf1 + NC;
    float* buf3  = buf2 + NC;
    float* dinv  = buf3 + NC;
    float* stats = dinv + N;

    auto zero = [&](float* p, size_t n) {
        unsigned g = (unsigned)((n + 255) / 256);
        if (g > 32768u) g = 32768u;
        fill_kernel<<<dim3(g), dim3(256), 0, stream>>>(p, 0.0f, (long)n);
    };
    auto gemm = [&](const float* X, int ldx, int K, const float* W, const float* bias,
                    int ncols, float* OUT, int ldout, int coff) {
        dim3 grid((N + 63) / 64, (ncols + 15) / 16);
        gemm_wmma_kernel<<<grid, dim3(128), 0, stream>>>(X, ldx, K, W, bias, ncols,
                                                         OUT, ldout, coff, N);
    };
    auto prop = [&](const float* in, float* o, int F) {
        zero(o, (size_t)N * F);
        prop_kernel<<<dim3(2048), dim3(256), 0, stream>>>(in, esrc, edst, dinv, o, F, E, N);
    };
    auto bnorm = [&](float* h, const float* g, const float* b) {
        zero(stats, 2 * CATF);
        bn_stats_kernel<<<dim3(1024), dim3(256), 0, stream>>>(h, stats, N, CATF);
        unsigned gr = (unsigned)(((size_t)N * CATF + 255) / 256);
        if (gr > 32768u) gr = 32768u;
        bn_apply_kernel<<<dim3(gr), dim3(256), 0, stream>>>(h, stats, g, b, N, CATF);
    };

    // --- gcn_norm: deg = in-degree + 1 (self loop); dinv = rsqrt(deg) ---
    {
        unsigned g = (unsigned)((N + 255) / 256);
        fill_kernel<<<dim3(g), dim3(256), 0, stream>>>(dinv, 1.0f, (long)N);
        deg_kernel<<<dim3(2048), dim3(256), 0, stream>>>(edst, dinv, E);
        rsqrt_kernel<<<dim3(g), dim3(256), 0, stream>>>(dinv, N);
    }

    // --- Layer 1 (input x: [N,128]) -> buf0 [N,180] ---
    gemm(x, F_IN, F_IN, c1_w[0], c1_b[0], HID, buf0, CATF, 0);
    prop(x, buf1, F_IN);
    gemm(buf1, F_IN, F_IN, c1_w[1], c1_b[1], HID, buf0, CATF, HID);
    prop(buf1, buf2, F_IN);
    gemm(buf2, F_IN, F_IN, c1_w[2], c1_b[2], HID, buf0, CATF, 2 * HID);
    bnorm(buf0, bn_g[0], bn_b[0]);

    // --- Layer 2 (input buf0: [N,180]) -> buf3 ---
    gemm(buf0, CATF, CATF, c2_w[0], c2_b[0], HID, buf3, CATF, 0);
    prop(buf0, buf1, CATF);
    gemm(buf1, CATF, CATF, c2_w[1], c2_b[1], HID, buf3, CATF, HID);
    prop(buf1, buf2, CATF);
    gemm(buf2, CATF, CATF, c2_w[2], c2_b[2], HID, buf3, CATF, 2 * HID);
    bnorm(buf3, bn_g[1], bn_b[1]);

    // --- Layer 3 (input buf3) -> buf0 ---
    gemm(buf3, CATF, CATF, c3_w[0], c3_b[0], HID, buf0, CATF, 0);
    prop(buf3, buf1, CATF);
    gemm(buf1, CATF, CATF, c3_w[1], c3_b[1], HID, buf0, CATF, HID);
    prop(buf1, buf2, CATF);
    gemm(buf2, CATF, CATF, c3_w[2], c3_b[2], HID, buf0, CATF, 2 * HID);
    bnorm(buf0, bn_g[2], bn_b[2]);

    // --- Final classifier: out[N,7] = buf0 @ lin_w + lin_b ---
    gemm(buf0, CATF, CATF, lin_w, lin_b, NCLS, out, NCLS, 0);
}